// Graph_ReLu_W_30502857736237
// MI455X (gfx1250) — compile-verified
//
#include <hip/hip_runtime.h>
#include <hip/hip_bf16.h>

#ifndef __has_builtin
#define __has_builtin(x) 0
#endif

#define N_NODES 8192
#define TOPK    32
#define BLOCK   256
#define NWAVES  (BLOCK / 32)
#define CHUNKS  8                 // 8 x vec4 per thread -> 32 elements/thread

typedef float v4f __attribute__((ext_vector_type(4)));
typedef int   v4i __attribute__((vector_size(16)));   // matches builtin's int4 pointee

// gfx1250 async global->LDS copy path (probe via __has_builtin; sync fallback otherwise)
#if defined(__AMDGCN__) && __has_builtin(__builtin_amdgcn_global_load_async_to_lds_b128)
#define USE_ASYNC 1
#else
#define USE_ASYNC 0
#endif

__global__ __launch_bounds__(BLOCK) void topk_mask_kernel(
    const float* __restrict__ A, const float* __restrict__ Nz, float* __restrict__ Out)
{
    const int tid  = threadIdx.x;
    const int wave = tid >> 5;                       // wave32
    const size_t base = (size_t)blockIdx.x * N_NODES;

    __shared__ unsigned s_hist[NWAVES][256];         // per-wave sub-histograms (8 KB)
    __shared__ unsigned s_suf[256];                  // suffix sums
    __shared__ unsigned s_selBin;
    __shared__ unsigned s_selRem;
    __shared__ unsigned s_eq;
#if USE_ASYNC
    __shared__ v4f s_Astage[N_NODES / 4];            // 32 KB staging for one row of A
#endif

    float    adj[CHUNKS * 4];
    unsigned pb [CHUNKS * 4];

#if USE_ASYNC
    // Kick off async copies of this row of A into LDS (each lane stages its own 8x16B).
    #pragma unroll
    for (int c = 0; c < CHUNKS; ++c) {
        const int slot = c * BLOCK + tid;            // v4f index
        const size_t g = base + (size_t)slot * 4;    // float index
        __builtin_amdgcn_global_load_async_to_lds_b128(
            (__attribute__((address_space(1))) v4i*)(A + g),
            (__attribute__((address_space(3))) v4i*)(&s_Astage[slot]),
            0, 0);
    }
#endif

    // Noise via plain non-temporal vector loads (overlaps with async A transfer).
    v4f nz[CHUNKS];
    #pragma unroll
    for (int c = 0; c < CHUNKS; ++c) {
        const int slot = c * BLOCK + tid;
        nz[c] = __builtin_nontemporal_load((const v4f*)(Nz + base + (size_t)slot * 4));
    }

#if USE_ASYNC
#if __has_builtin(__builtin_amdgcn_s_wait_asynccnt)
    __builtin_amdgcn_s_wait_asynccnt(0);
    asm volatile("" ::: "memory");
#else
    asm volatile("s_wait_asynccnt 0" ::: "memory");
#endif
#endif

    #pragma unroll
    for (int c = 0; c < CHUNKS; ++c) {
        const int slot = c * BLOCK + tid;
#if USE_ASYNC
        v4f a = s_Astage[slot];                      // lane reads only its own staged data
#else
        v4f a = __builtin_nontemporal_load((const v4f*)(A + base + (size_t)slot * 4));
#endif
        #pragma unroll
        for (int v = 0; v < 4; ++v) {
            float ad = a[v] > 0.0f ? a[v] : 0.0f;    // relu
            adj[c * 4 + v] = ad;
            float p = fmaf(nz[c][v], 0.01f, ad);     // perturbed; >= 0 so bits are order-preserving
            pb[c * 4 + v] = __float_as_uint(p);
        }
    }

    // ---- 4-round radix select (8 bits/round, MSB first) for the K-th largest value ----
    unsigned prefix = 0, pmask = 0;
    unsigned rem = TOPK;

    for (int shift = 24; shift >= 0; shift -= 8) {
        #pragma unroll
        for (int w = 0; w < NWAVES; ++w) s_hist[w][tid] = 0;
        __syncthreads();

        #pragma unroll
        for (int e = 0; e < CHUNKS * 4; ++e) {
            if ((pb[e] & pmask) == prefix)
                atomicAdd(&s_hist[wave][(pb[e] >> shift) & 0xFFu], 1u);
        }
        __syncthreads();

        unsigned tot = 0;
        #pragma unroll
        for (int w = 0; w < NWAVES; ++w) tot += s_hist[w][tid];
        s_suf[tid] = tot;
        __syncthreads();

        // Hillis-Steele suffix (reverse) inclusive scan over 256 bins
        for (int st = 1; st < 256; st <<= 1) {
            unsigned t = s_suf[tid];
            if (tid + st < 256) t += s_suf[tid + st];
            __syncthreads();
            s_suf[tid] = t;
            __syncthreads();
        }

        unsigned here = s_suf[tid];
        unsigned nxt  = (tid < 255) ? s_suf[tid + 1] : 0u;
        if (here >= rem && nxt < rem) {              // unique by monotonicity
            s_selBin = (unsigned)tid;
            s_selRem = rem - nxt;
        }
        __syncthreads();

        prefix |= s_selBin << shift;
        pmask  |= 0xFFu << shift;
        rem     = s_selRem;
        __syncthreads();
    }

    // prefix == bit pattern of the K-th largest perturbed value; rem == how many
    // threshold-equal elements to keep (exactly K total).
    if (tid == 0) s_eq = 0;
    __syncthreads();

    const unsigned T     = prefix;
    const unsigned quota = rem;

    #pragma unroll
    for (int c = 0; c < CHUNKS; ++c) {
        const int slot = c * BLOCK + tid;
        v4f o;
        #pragma unroll
        for (int v = 0; v < 4; ++v) {
            const int e = c * 4 + v;
            float r = 0.0f;
            if (pb[e] > T) {
                r = adj[e];
            } else if (pb[e] == T) {
                unsigned pos = atomicAdd(&s_eq, 1u);
                if (pos < quota) r = adj[e];
            }
            o[v] = r;
        }
        __builtin_nontemporal_store(o, (v4f*)(Out + base + (size_t)slot * 4));
    }
}

extern "C" void kernel_launch(void* const* d_in, const int* in_sizes, int n_in,
                              void* d_out, int out_size, void* d_ws, size_t ws_size,
                              hipStream_t stream) {
    (void)in_sizes; (void)n_in; (void)out_size; (void)d_ws; (void)ws_size;
    // setup_inputs order: 0 = idx (int64, unused), 1 = A (f32 N*N), 2 = noise (f32 N*N)
    const float* A  = (const float*)d_in[1];
    const float* Nz = (const float*)d_in[2];
    float* Out      = (float*)d_out;

    dim3 grid(N_NODES), block(BLOCK);
    hipLaunchKernelGGL(topk_mask_kernel, grid, block, 0, stream, A, Nz, Out);
}